// MaxPoolingLayerMatch_34986803593512
// MI455X (gfx1250) — compile-verified
//
#include <hip/hip_runtime.h>
#include <math.h>

// MI455X / gfx1250, wave32. Store-bound problem (335MB output @23.3TB/s ~ 14.4us),
// f32 WMMA (V_WMMA_F32_16X16X4_F32) for the 512x512x128 cosine GEMMs.

typedef __attribute__((ext_vector_type(2))) float v2f;
typedef __attribute__((ext_vector_type(8))) float v8f;

#define BB   16
#define S1D  512
#define S2D  512
#define HD   128
#define LD   20
#define EPSV 1e-8f
#define PADW 132   // 16x128 tile rows padded by 4 floats -> conflict-free b64 LDS reads

// ---------------------------------------------------------------------------
// Kernel 1: per-perspective norms  n[b,l,s] = || w[l,:] * v[b,s,:] ||
// grid = (512, 16, 2), block = 64 (2 waves). Tiny kernel, negligible cost.
// ---------------------------------------------------------------------------
__global__ __launch_bounds__(64)
void mpm_norms_kernel(const float* __restrict__ v1,
                      const float* __restrict__ v2,
                      const float* __restrict__ w,
                      float* __restrict__ n1,
                      float* __restrict__ n2) {
    __shared__ float vsq[HD];
    const int s     = blockIdx.x;
    const int b     = blockIdx.y;
    const int which = blockIdx.z;
    const float* v  = (which == 0) ? v1 : v2;
    float* nout     = (which == 0) ? n1 : n2;
    const int tid   = threadIdx.x;

    for (int h = tid; h < HD; h += 64) {
        float x = v[((size_t)b * S1D + s) * HD + h];   // S1D == S2D
        vsq[h] = x * x;
    }
    __syncthreads();

    if (tid < LD) {
        const float* wl = w + tid * HD;
        float sum = 0.f;
        #pragma unroll 4
        for (int h = 0; h < HD; ++h) {
            float ww = wl[h];
            sum = fmaf(ww * ww, vsq[h], sum);          // (w*v)^2 = w^2 * v^2
        }
        nout[((size_t)b * LD + tid) * S1D + s] = sqrtf(sum);
    }
}

// ---------------------------------------------------------------------------
// Kernel 2: main match kernel.
// grid = (S2/16, S1/16, B), block = 128 (4 wave32). Each wave handles 5 of the
// 20 perspectives; per perspective: 16x16x128 GEMM tile via 32x
// v_wmma_f32_16x16x4_f32, w[l,h]^2 (pre-squared at LDS staging) folded into the
// A operand on load -> single v_pk_mul_f32 feeding each WMMA. Results staged in
// LDS in (s,t,l) output order, then flushed with coalesced float4 stores.
// ---------------------------------------------------------------------------
__global__ __launch_bounds__(128)
void mpm_match_kernel(const float* __restrict__ v1,
                      const float* __restrict__ v2,
                      const float* __restrict__ w,
                      const float* __restrict__ n1,
                      const float* __restrict__ n2,
                      float* __restrict__ out) {
    __shared__ float As[16 * PADW];                    // v1 tile (raw)
    __shared__ float Bs[16 * PADW];                    // v2 tile (raw)
    __shared__ float Ws[LD * HD];                      // w^2, all 20 perspectives
    __shared__ __align__(16) float Cs[16 * 16 * LD];   // output tile (s,t,l)

    const int t0  = blockIdx.x * 16;
    const int s0  = blockIdx.y * 16;
    const int b   = blockIdx.z;
    const int tid = threadIdx.x;                       // 0..127 == H

    // Stage tiles: thread tid loads column h=tid of each of the 16 rows (coalesced 512B).
    #pragma unroll
    for (int j = 0; j < 16; ++j) {
        As[j * PADW + tid] = v1[((size_t)b * S1D + s0 + j) * HD + tid];
        Bs[j * PADW + tid] = v2[((size_t)b * S2D + t0 + j) * HD + tid];
    }
    for (int i = tid; i < LD * HD; i += 128) {
        float ww = w[i];
        Ws[i] = ww * ww;                               // pre-square: hoists one pk_mul
    }
    __syncthreads();

    const int wave = tid >> 5;        // 0..3 (uniform per wave)
    const int lane = tid & 31;
    const int mn   = lane & 15;       // row of A frag == col of B frag for this lane
    const int hi   = lane >> 4;       // half-wave select
    const int k0   = hi << 1;         // K sub-offset {0,2} per ISA f32 A/B layout

    for (int li = 0; li < 5; ++li) {
        const int l = wave * 5 + li;  // uniform per wave -> EXEC stays all-ones
        const float* wrow = &Ws[l * HD];
        v8f acc = {0.f, 0.f, 0.f, 0.f, 0.f, 0.f, 0.f, 0.f};

        #pragma unroll 8
        for (int ko = 0; ko < HD; ko += 4) {
            v2f a, bmat;
            a.x    = wrow[ko + k0]     * As[mn * PADW + ko + k0];
            a.y    = wrow[ko + k0 + 1] * As[mn * PADW + ko + k0 + 1];
            bmat.x = Bs[mn * PADW + ko + k0];
            bmat.y = Bs[mn * PADW + ko + k0 + 1];
            acc = __builtin_amdgcn_wmma_f32_16x16x4_f32(
                /*neg_a=*/false, a, /*neg_b=*/false, bmat,
                /*c_mod=*/(short)0, acc, /*reuse_a=*/false, /*reuse_b=*/false);
        }

        // Apply cosine denominator with the reference's eps rule, stash into LDS.
        const float  n2v = n2[((size_t)b * LD + l) * S2D + t0 + mn];  // per-lane col norm
        const float* n1p = &n1[((size_t)b * LD + l) * S1D + s0];
        #pragma unroll
        for (int r = 0; r < 8; ++r) {
            const int   m   = r + (hi << 3);           // C layout: VGPR r -> rows r / r+8
            const float d   = n1p[m] * n2v;
            const float den = (d > EPSV) ? d : EPSV;
            Cs[(m * 16 + mn) * LD + l] = acc[r] / den;
        }
    }
    __syncthreads();

    // Coalesced flush: 16*16*20 = 5120 floats = 1280 float4; 10 per thread.
    // Each s-row of the tile is 16*20 = 320 contiguous floats in global memory.
    const float4* cs4 = (const float4*)Cs;
    #pragma unroll
    for (int j = 0; j < 10; ++j) {
        const int idx4 = j * 128 + tid;
        const int flat = idx4 * 4;
        const int srow = flat / (16 * LD);
        const int rem  = flat - srow * (16 * LD);
        float* dst = out + ((((size_t)b * S1D + s0 + srow) * S2D + t0) * LD + rem);
        *(float4*)dst = cs4[idx4];                     // 16B-aligned (offsets %4 == 0)
    }
}

// ---------------------------------------------------------------------------
extern "C" void kernel_launch(void* const* d_in, const int* in_sizes, int n_in,
                              void* d_out, int out_size, void* d_ws, size_t ws_size,
                              hipStream_t stream) {
    const float* v1 = (const float*)d_in[0];   // (16,512,128) f32
    const float* v2 = (const float*)d_in[1];   // (16,512,128) f32
    const float* w  = (const float*)d_in[2];   // (20,128)     f32
    float* out = (float*)d_out;                // (16,512,512,20) f32

    float* n1 = (float*)d_ws;                  // B*L*S1 floats
    float* n2 = n1 + (size_t)BB * LD * S1D;    // B*L*S2 floats (total ~1.3MB of ws)

    mpm_norms_kernel<<<dim3(S1D, BB, 2), 64, 0, stream>>>(v1, v2, w, n1, n2);
    mpm_match_kernel<<<dim3(S2D / 16, S1D / 16, BB), 128, 0, stream>>>(
        v1, v2, w, n1, n2, out);
}